// Indexing_Blend_Kron3_MLP_5557687681037
// MI455X (gfx1250) — compile-verified
//
#include <hip/hip_runtime.h>
#include <hip/hip_bf16.h>

typedef __attribute__((ext_vector_type(2))) float v2f;
typedef __attribute__((ext_vector_type(8))) float v8f;

#define NGRID 160
#define DDIM  48
#define NPTS  2000000
#define KNN   8
#define NVOX  (160*160*160)

// ---- per-wave 16x16 f32 WMMA tile: C = A(m0:,:K) * B(:K, n0:) -------------
// A row-major (lda), B either row-major K x N (btrans=false) or stored as
// N x K row-major accessed transposed (btrans=true).
__device__ __forceinline__ v8f wmma_tile(const float* __restrict__ A, int lda,
                                         const float* __restrict__ B, int ldb,
                                         bool btrans, int K, int m0, int n0) {
  const int lane = threadIdx.x & 31;
  const int half = lane >> 4;   // 0: lanes 0-15, 1: lanes 16-31
  const int ln   = lane & 15;
  v8f acc = {};
  const float* arow = A + (size_t)(m0 + ln) * lda;
  for (int k0 = 0; k0 < K; k0 += 4) {
    const int ka = k0 + half * 2;
    v2f a, b;
    a[0] = arow[ka];
    a[1] = arow[ka + 1];
    if (btrans) {
      const float* brow = B + (size_t)(n0 + ln) * ldb;
      b[0] = brow[ka];
      b[1] = brow[ka + 1];
    } else {
      b[0] = B[(size_t)ka * ldb + (n0 + ln)];
      b[1] = B[(size_t)(ka + 1) * ldb + (n0 + ln)];
    }
    // D = A*B + C   (16x16x4, full f32)
    acc = __builtin_amdgcn_wmma_f32_16x16x4_f32(false, a, false, b,
                                                (short)0, acc, false, false);
  }
  return acc;
}

// ---- K1: T1[w] (160 x 2304) = x (160x48) @ W[w] (48 x 2304) ---------------
__global__ void k1_xw(const float* __restrict__ x, const float* __restrict__ W,
                      float* __restrict__ T1) {
  const int w    = blockIdx.z;
  const int wave = threadIdx.x >> 5;
  const int tile = blockIdx.x * (blockDim.x >> 5) + wave;   // 0..1439
  const int Nt   = (DDIM * DDIM) / 16;                      // 144
  const int tm   = tile / Nt, tn = tile % Nt;
  const float* Bm = W  + (size_t)w * DDIM * DDIM * DDIM;    // 48 x 2304
  float*       C  = T1 + (size_t)w * NGRID * DDIM * DDIM;   // 160 x 2304
  v8f acc = wmma_tile(x, DDIM, Bm, DDIM * DDIM, false, DDIM, tm * 16, tn * 16);
  const int lane = threadIdx.x & 31, half = lane >> 4, ln = lane & 15;
#pragma unroll
  for (int v = 0; v < 8; ++v)
    C[(size_t)(tm * 16 + half * 8 + v) * (DDIM * DDIM) + tn * 16 + ln] = acc[v];
}

// ---- K2: g2[w,a] (160 x 48) = y (160x48) @ T1[w,a] (48 x 48) --------------
__global__ void k2_y(const float* __restrict__ y, const float* __restrict__ T1,
                     float* __restrict__ g2) {
  const int tm = blockIdx.x;   // 0..9
  const int tn = blockIdx.y;   // 0..2
  const int wa = blockIdx.z;   // 0..479
  const float* Bm = T1 + (size_t)wa * DDIM * DDIM;
  float*       C  = g2 + (size_t)wa * NGRID * DDIM;
  v8f acc = wmma_tile(y, DDIM, Bm, DDIM, false, DDIM, tm * 16, tn * 16);
  const int lane = threadIdx.x & 31, half = lane >> 4, ln = lane & 15;
#pragma unroll
  for (int v = 0; v < 8; ++v)
    C[(size_t)(tm * 16 + half * 8 + v) * DDIM + tn * 16 + ln] = acc[v];
}

// ---- K3: g (76800 x 160) = g2 (76800 x 48) @ z^T (48 x 160),
//          scattered to interleaved g_i[(spatial)*3 + w] ---------------------
__global__ void k3_z(const float* __restrict__ g2, const float* __restrict__ z,
                     float* __restrict__ gi) {
  const int wave = threadIdx.x >> 5;
  const int tile = blockIdx.x * (blockDim.x >> 5) + wave;   // 0..47999
  const int Nt   = NGRID / 16;                              // 10
  const int tm   = tile / Nt, tn = tile % Nt;
  v8f acc = wmma_tile(g2, DDIM, z, DDIM, true, DDIM, tm * 16, tn * 16);
  const int lane = threadIdx.x & 31, half = lane >> 4, ln = lane & 15;
#pragma unroll
  for (int v = 0; v < 8; ++v) {
    const int row = tm * 16 + half * 8 + v;   // (w*160 + a)*160 + b
    const int col = tn * 16 + ln;             // c
    const int w   = row / (NGRID * NGRID);
    const int ab  = row % (NGRID * NGRID);
    const size_t i = (size_t)ab * NGRID + col;
    gi[i * 3 + w] = acc[v];
  }
}

// ---- zero the regularizer accumulator slot --------------------------------
__global__ void k_zero(float* p) { *p = 0.0f; }

// ---- K4: total-variation term, LDS tree reduce + one atomic per block -----
__global__ void k4_tv(const float* __restrict__ gi, float* __restrict__ reg_out) {
  __shared__ float sred[256];
  const int n1 = NGRID - 1;                                 // 159
  const long long t = (long long)blockIdx.x * blockDim.x + threadIdx.x;
  float val = 0.0f;
  if (t < (long long)n1 * n1 * n1) {
    const int c = (int)(t % n1);
    const int b = (int)((t / n1) % n1);
    const int a = (int)(t / ((long long)n1 * n1));
    const size_t i = ((size_t)a * NGRID + b) * NGRID + c;
    const float* p  = gi + i * 3;
    const float* px = p + 3;                                // (a,b,c+1)
    const float* py = p + (size_t)NGRID * 3;                // (a,b+1,c)
    const float* pz = p + (size_t)NGRID * NGRID * 3;        // (a+1,b,c)
#pragma unroll
    for (int w = 0; w < 3; ++w) {
      const float base = p[w];
      const float dx = px[w] - base;
      const float dy = py[w] - base;
      const float dz = pz[w] - base;
      val += sqrtf(1e-5f + dx * dx + dy * dy + dz * dz);
    }
  }
  sred[threadIdx.x] = val;
  __syncthreads();
  for (int s = 128; s > 0; s >>= 1) {
    if (threadIdx.x < s) sred[threadIdx.x] += sred[threadIdx.x + s];
    __syncthreads();
  }
  if (threadIdx.x == 0) {
    const float scale = 1.0f / ((float)n1 * (float)n1 * (float)n1);  // mean
    atomicAdd(reg_out, sred[0] * scale);
  }
}

// ---- K5: weighted 8-NN gather; g_i interleaved -> 12B contiguous per tap --
__global__ void k5_gather(const int* __restrict__ B_idx,
                          const float* __restrict__ B_w,
                          const float* __restrict__ gi,
                          float* __restrict__ out) {
  const int n = blockIdx.x * blockDim.x + threadIdx.x;
  if (n >= NPTS) return;
  const int*   idx = B_idx + (size_t)n * KNN;
  const float* wv  = B_w   + (size_t)n * KNN;
  float o0 = 0.0f, o1 = 0.0f, o2 = 0.0f;
#pragma unroll
  for (int k = 0; k < KNN; ++k) {
    const float* p  = gi + (size_t)idx[k] * 3;
    const float  wk = wv[k];
    o0 += wk * p[0];
    o1 += wk * p[1];
    o2 += wk * p[2];
  }
  out[(size_t)n * 3 + 0] = o0;
  out[(size_t)n * 3 + 1] = o1;
  out[(size_t)n * 3 + 2] = o2;
}

extern "C" void kernel_launch(void* const* d_in, const int* in_sizes, int n_in,
                              void* d_out, int out_size, void* d_ws, size_t ws_size,
                              hipStream_t stream) {
  const float* x     = (const float*)d_in[0];
  const float* y     = (const float*)d_in[1];
  const float* z     = (const float*)d_in[2];
  const float* W     = (const float*)d_in[3];
  const int*   B_idx = (const int*)d_in[4];
  const float* B_w   = (const float*)d_in[5];
  float* out = (float*)d_out;

  char* ws = (char*)d_ws;
  float* gi = (float*)ws;                                       // 49,152,000 B
  float* T1 = (float*)(ws + (size_t)NVOX * 3 * 4);              //  4,423,680 B
  float* g2 = (float*)(ws + (size_t)NVOX * 3 * 4
                          + (size_t)3 * NGRID * DDIM * DDIM * 4); // 14,745,600 B

  // Stage 1: 3 x (10 x 144) tiles, 8 waves/block
  k1_xw<<<dim3(180, 1, 3), 256, 0, stream>>>(x, W, T1);
  // Stage 2: 480 batched 160x48x48 GEMMs, 1 wave per 16x16 tile
  k2_y<<<dim3(10, 3, 480), 32, 0, stream>>>(y, T1, g2);
  // Stage 3: 76800 x 160 x 48 GEMM, scatter interleaved
  k3_z<<<dim3(6000, 1, 1), 256, 0, stream>>>(g2, z, gi);
  // Regularizer (zero slot, then reduce)
  k_zero<<<1, 1, 0, stream>>>(out + (size_t)NPTS * 3);
  const long long nvtv = 159LL * 159 * 159;
  k4_tv<<<(unsigned)((nvtv + 255) / 256), 256, 0, stream>>>(gi, out + (size_t)NPTS * 3);
  // Gather (the bandwidth-dominant stage)
  k5_gather<<<(NPTS + 255) / 256, 256, 0, stream>>>(B_idx, B_w, gi, out);
}